// SensorGCN_24429773980016
// MI455X (gfx1250) — compile-verified
//
#include <hip/hip_runtime.h>
#include <math.h>

#define N_NODES 100000
#define N_EDGES 2500000
#define N_GRAPHS 1000
#define F_INDIM 4
#define HDIM 32
#define CDIM 3

typedef __attribute__((ext_vector_type(2))) float v2f;
typedef __attribute__((ext_vector_type(8))) float v8f;

// ---------------- degree / normalization ----------------

__global__ void init_deg_kernel(float* __restrict__ deg) {
    int n = blockIdx.x * blockDim.x + threadIdx.x;
    if (n < N_NODES) deg[n] = 1.0f;  // self-loop weight
}

__global__ void deg_accum_kernel(const long long* __restrict__ ei,
                                 const float* __restrict__ w,
                                 float* __restrict__ deg) {
    int e = blockIdx.x * blockDim.x + threadIdx.x;
    if (e < N_EDGES) {
        long long col = ei[(long long)N_EDGES + e];
        atomicAdd(&deg[col], w[e]);
    }
}

__global__ void dinv_kernel(float* __restrict__ deg) {
    int n = blockIdx.x * blockDim.x + threadIdx.x;
    if (n < N_NODES) {
        float d = deg[n];
        deg[n] = (d > 0.0f) ? rsqrtf(d) : 0.0f;
    }
}

// ---------------- dense matmul via V_WMMA_F32_16X16X4_F32 ----------------
// out[N,32] = act[N,K] @ W[K,32]; one wave handles a 16-row tile (two 16x16
// output tiles), accumulating over K in steps of 4.
template <int K>
__global__ void gcn_matmul_wmma(const float* __restrict__ act,
                                const float* __restrict__ W,
                                float* __restrict__ out) {
    int wave = (blockIdx.x * blockDim.x + threadIdx.x) >> 5;
    int lane = threadIdx.x & 31;
    int base = wave * 16;
    if (base >= N_NODES) return;  // uniform per wave -> EXEC stays all-1s

    int mn    = lane & 15;          // M for A, N for B/C/D
    int khalf = (lane >> 4) * 2;    // lanes 0-15: K={0,1}; lanes 16-31: K={2,3}

    v8f c0 = {};
    v8f c1 = {};
#pragma unroll
    for (int k0 = 0; k0 < K; k0 += 4) {
        int row = base + mn;
        v2f a;
        if (row < N_NODES) {
            a.x = act[(long long)row * K + k0 + khalf];
            a.y = act[(long long)row * K + k0 + khalf + 1];
        } else {
            a.x = 0.0f;
            a.y = 0.0f;
        }
        v2f b0, b1;
        b0.x = W[(k0 + khalf) * HDIM + mn];
        b0.y = W[(k0 + khalf + 1) * HDIM + mn];
        b1.x = W[(k0 + khalf) * HDIM + mn + 16];
        b1.y = W[(k0 + khalf + 1) * HDIM + mn + 16];

        c0 = __builtin_amdgcn_wmma_f32_16x16x4_f32(false, a, false, b0,
                                                   (short)0, c0, false, false);
        c1 = __builtin_amdgcn_wmma_f32_16x16x4_f32(false, a, false, b1,
                                                   (short)0, c1, false, false);
    }

    int mbase = (lane >> 4) * 8;
#pragma unroll
    for (int r = 0; r < 8; ++r) {
        int row = base + mbase + r;
        if (row < N_NODES) {
            out[(long long)row * HDIM + mn]      = c0[r];
            out[(long long)row * HDIM + mn + 16] = c1[r];
        }
    }
}

// ---------------- sparse aggregation ----------------

__global__ void zero_kernel(float* __restrict__ p, long long n) {
    long long i = (long long)blockIdx.x * blockDim.x + threadIdx.x;
    if (i < n) p[i] = 0.0f;
}

// one thread per (edge, feature): agg[col,f] += dinv[row]*w*dinv[col] * t[row,f]
__global__ void scatter_kernel(const long long* __restrict__ ei,
                               const float* __restrict__ w,
                               const float* __restrict__ dinv,
                               const float* __restrict__ t,
                               float* __restrict__ agg) {
    long long idx = (long long)blockIdx.x * blockDim.x + threadIdx.x;
    if (idx >= (long long)N_EDGES * HDIM) return;
    int e = (int)(idx >> 5);
    int f = (int)(idx & 31);
    long long row = ei[e];
    long long col = ei[(long long)N_EDGES + e];
    float norm = dinv[row] * w[e] * dinv[col];
    atomicAdd(&agg[col * HDIM + f], norm * t[row * HDIM + f]);
}

// self-loop contribution + bias (+ optional relu), written in place into agg
__global__ void finish_kernel(const float* __restrict__ t,
                              const float* __restrict__ dinv,
                              const float* __restrict__ bias,
                              float* __restrict__ agg, int do_relu) {
    long long idx = (long long)blockIdx.x * blockDim.x + threadIdx.x;
    if (idx >= (long long)N_NODES * HDIM) return;
    int n = (int)(idx >> 5);
    int f = (int)(idx & 31);
    float di = dinv[n];
    float v = agg[idx] + di * di * t[idx] + bias[f];
    agg[idx] = do_relu ? fmaxf(v, 0.0f) : v;
}

// ---------------- pooling + head ----------------

__global__ void pool_kernel(const float* __restrict__ h,
                            const long long* __restrict__ batch,
                            float* __restrict__ sums, float* __restrict__ cnt) {
    long long idx = (long long)blockIdx.x * blockDim.x + threadIdx.x;
    if (idx >= (long long)N_NODES * HDIM) return;
    int n = (int)(idx >> 5);
    int f = (int)(idx & 31);
    long long g = batch[n];
    atomicAdd(&sums[g * HDIM + f], h[idx]);
    if (f == 0) atomicAdd(&cnt[g], 1.0f);
}

__global__ void head_kernel(const float* __restrict__ sums,
                            const float* __restrict__ cnt,
                            const float* __restrict__ lin_w,
                            const float* __restrict__ lin_b,
                            float* __restrict__ out) {
    int g = blockIdx.x * blockDim.x + threadIdx.x;
    if (g >= N_GRAPHS) return;
    float inv = 1.0f / fmaxf(cnt[g], 1.0f);
    float logits[CDIM];
#pragma unroll
    for (int c = 0; c < CDIM; ++c) logits[c] = lin_b[c];
#pragma unroll
    for (int f = 0; f < HDIM; ++f) {
        float p = sums[g * HDIM + f] * inv;
#pragma unroll
        for (int c = 0; c < CDIM; ++c) logits[c] += p * lin_w[f * CDIM + c];
    }
    float mx = fmaxf(logits[0], fmaxf(logits[1], logits[2]));
    float ex[CDIM];
    float s = 0.0f;
#pragma unroll
    for (int c = 0; c < CDIM; ++c) { ex[c] = expf(logits[c] - mx); s += ex[c]; }
    float is = 1.0f / s;
#pragma unroll
    for (int c = 0; c < CDIM; ++c) out[g * CDIM + c] = ex[c] * is;
}

// ---------------- launch ----------------

static inline size_t align256(size_t x) { return (x + 255) & ~(size_t)255; }

extern "C" void kernel_launch(void* const* d_in, const int* in_sizes, int n_in,
                              void* d_out, int out_size, void* d_ws, size_t ws_size,
                              hipStream_t stream) {
    const float*     x     = (const float*)d_in[0];
    const long long* ei    = (const long long*)d_in[1];  // int64 [2, E]
    const float*     ew    = (const float*)d_in[2];
    const long long* batch = (const long long*)d_in[3];  // int64 [N]
    const float*     W1    = (const float*)d_in[4];
    const float*     b1    = (const float*)d_in[5];
    const float*     W2    = (const float*)d_in[6];
    const float*     b2    = (const float*)d_in[7];
    const float*     W3    = (const float*)d_in[8];
    const float*     b3    = (const float*)d_in[9];
    const float*     lin_w = (const float*)d_in[10];
    const float*     lin_b = (const float*)d_in[11];
    float*           out   = (float*)d_out;

    char* ws = (char*)d_ws;
    size_t off = 0;
    float* dinv = (float*)(ws + off); off = align256(off + sizeof(float) * N_NODES);
    float* bufA = (float*)(ws + off); off = align256(off + sizeof(float) * (size_t)N_NODES * HDIM);
    float* bufB = (float*)(ws + off); off = align256(off + sizeof(float) * (size_t)N_NODES * HDIM);
    float* sums = (float*)(ws + off); off = align256(off + sizeof(float) * (size_t)N_GRAPHS * HDIM);
    float* cnt  = (float*)(ws + off); off = align256(off + sizeof(float) * N_GRAPHS);
    (void)ws_size; (void)n_in; (void)in_sizes; (void)out_size;

    const int T = 256;
    const int gN  = (N_NODES + T - 1) / T;
    const int gE  = (N_EDGES + T - 1) / T;
    const long long nfTotal = (long long)N_NODES * HDIM;
    const int gNF = (int)((nfTotal + T - 1) / T);
    const long long efTotal = (long long)N_EDGES * HDIM;
    const int gEF = (int)((efTotal + T - 1) / T);
    const int tiles  = (N_NODES + 15) / 16;     // 16-row WMMA tiles
    const int gMM    = (tiles + (T / 32) - 1) / (T / 32);  // 8 waves / block

    // normalization: deg = 1 + segment_sum(w, col); dinv = rsqrt(deg)
    init_deg_kernel<<<gN, T, 0, stream>>>(dinv);
    deg_accum_kernel<<<gE, T, 0, stream>>>(ei, ew, dinv);
    dinv_kernel<<<gN, T, 0, stream>>>(dinv);

    // ---- layer 1: x[N,4] @ W1 -> bufA; scatter -> bufB; relu ----
    gcn_matmul_wmma<F_INDIM><<<gMM, T, 0, stream>>>(x, W1, bufA);
    zero_kernel<<<gNF, T, 0, stream>>>(bufB, nfTotal);
    scatter_kernel<<<gEF, T, 0, stream>>>(ei, ew, dinv, bufA, bufB);
    finish_kernel<<<gNF, T, 0, stream>>>(bufA, dinv, b1, bufB, 1);

    // ---- layer 2: bufB[N,32] @ W2 -> bufA; scatter -> bufB; relu ----
    gcn_matmul_wmma<HDIM><<<gMM, T, 0, stream>>>(bufB, W2, bufA);
    zero_kernel<<<gNF, T, 0, stream>>>(bufB, nfTotal);
    scatter_kernel<<<gEF, T, 0, stream>>>(ei, ew, dinv, bufA, bufB);
    finish_kernel<<<gNF, T, 0, stream>>>(bufA, dinv, b2, bufB, 1);

    // ---- layer 3: no relu ----
    gcn_matmul_wmma<HDIM><<<gMM, T, 0, stream>>>(bufB, W3, bufA);
    zero_kernel<<<gNF, T, 0, stream>>>(bufB, nfTotal);
    scatter_kernel<<<gEF, T, 0, stream>>>(ei, ew, dinv, bufA, bufB);
    finish_kernel<<<gNF, T, 0, stream>>>(bufA, dinv, b3, bufB, 0);

    // ---- mean pool + linear + softmax ----
    zero_kernel<<<(N_GRAPHS * HDIM + N_GRAPHS + T - 1) / T, T, 0, stream>>>(
        sums, (long long)N_GRAPHS * HDIM + N_GRAPHS);  // sums and cnt contiguous
    pool_kernel<<<gNF, T, 0, stream>>>(bufB, batch, sums, cnt);
    head_kernel<<<(N_GRAPHS + T - 1) / T, T, 0, stream>>>(sums, cnt, lin_w, lin_b, out);
}